// DecodePredictions_8366596293243
// MI455X (gfx1250) — compile-verified
//
#include <hip/hip_runtime.h>
#include <stdint.h>

// ---------------------------------------------------------------------------
// DecodePredictions + greedy NMS for MI455X (gfx1250)
//   Stage 1: decode 512x512x100 fp32 via async global->LDS tile staging
//            (25 unrolled global_load_async_to_lds_b128 per thread-slot)
//   Stage 2: 100 x (fused suppress+argmax -> select), L2-resident (10 MB)
// ---------------------------------------------------------------------------

#define G            512
#define CELLS        (G * G)        // 262144
#define NBOX         (CELLS * 2)    // 524288
#define NUM_CLASSES  90
#define CH           100
#define MAX_OUT      100
#define NEG30        (-1.0e30f)
#define TILE_CELLS   128

// ---------------- gfx1250 async global -> LDS helpers ----------------------
typedef int v4i_t __attribute__((ext_vector_type(4)));
typedef __attribute__((address_space(1))) v4i_t gv4i_t;   // global int4
typedef __attribute__((address_space(3))) v4i_t lv4i_t;   // LDS int4

__device__ __forceinline__ void async_copy_b128(const float* __restrict__ gsrc,
                                                float* ldst) {
#if defined(__gfx1250__) && __has_builtin(__builtin_amdgcn_global_load_async_to_lds_b128)
  // AS1 pointer == 64-bit generic address for global memory; AS3 pointer ==
  // low 32 bits of the generic LDS pointer (LDS byte offset).
  __builtin_amdgcn_global_load_async_to_lds_b128(
      (gv4i_t*)(uintptr_t)gsrc,
      (lv4i_t*)(uint32_t)(uintptr_t)ldst,
      /*offset=*/0, /*cpol=*/0);
#else
  // ISA 15.18.3 op 98: VDST = LDS byte offset VGPR, VADDR = 64-bit address.
  asm volatile("global_load_async_to_lds_b128 %0, %1, off"
               :
               : "v"((uint32_t)(uintptr_t)ldst),
                 "v"((unsigned long long)(uintptr_t)gsrc)
               : "memory");
#endif
}

__device__ __forceinline__ void wait_async0() {
#if defined(__gfx1250__) && __has_builtin(__builtin_amdgcn_s_wait_asynccnt)
  __builtin_amdgcn_s_wait_asynccnt(0);
#else
  asm volatile("s_wait_asynccnt 0" ::: "memory");
#endif
}

// ---------------- Stage 1: decode ------------------------------------------
__global__ __launch_bounds__(TILE_CELLS)
void decode_kernel(const float* __restrict__ in, const int* __restrict__ sq_flag,
                   float4* __restrict__ boxes, float* __restrict__ live,
                   int* __restrict__ cls, float* __restrict__ hdr) {
  __shared__ float tile[TILE_CELLS * CH];   // 51200 B

  const int cell0 = blockIdx.x * TILE_CELLS;
  const float* gbase = in + (size_t)cell0 * CH;

  // Stage 128 cells * 100 ch = 3200 x 16B chunks; constant trip count (25)
  // so the compiler emits a straight run of async loads + one wait.
#pragma unroll
  for (int it = 0; it < (TILE_CELLS * CH / 4) / TILE_CELLS; ++it) {  // 25
    const int k = it * TILE_CELLS + threadIdx.x;
    async_copy_b128(gbase + 4 * k, &tile[4 * k]);
  }
  wait_async0();
  __syncthreads();

  // initialize "no selected box yet" for the first fused NMS pass
  if (blockIdx.x == 0 && threadIdx.x == 0) hdr[5] = 0.0f;

  const float* t = &tile[threadIdx.x * CH];
  const int cell = cell0 + threadIdx.x;
  const float fj = (float)(cell & (G - 1));   // column (mesh X)
  const float fi = (float)(cell >> 9);        // row    (mesh Y)

  // argmax over 90 classes, first-index tie-break (strict >)
  float best = t[10];
  int bid = 0;
  for (int c = 1; c < NUM_CLASSES; ++c) {
    float v = t[10 + c];
    if (v > best) { best = v; bid = c; }
  }

  const int sq = sq_flag[0];

#pragma unroll
  for (int a = 0; a < 2; ++a) {
    float cx = (t[2 + 4 * a] + fj) * 16.0f;
    float cy = (t[3 + 4 * a] + fi) * 16.0f;
    float w = t[4 + 4 * a];
    float h = t[5 + 4 * a];
    if (sq) { w = w * w; h = h * h; }
    w *= 8192.0f;
    h *= 8192.0f;
    float x1 = cx - w * 0.5f;
    float y1 = cy - h * 0.5f;
    float x2 = cx + w * 0.5f - 1.0f;
    float y2 = cy + h * 0.5f - 1.0f;
    float score = best * t[a];
    int idx = cell * 2 + a;
    boxes[idx] = make_float4(x1, y1, x2, y2);
    live[idx] = (score > 0.6f) ? score : NEG30;
    cls[idx] = bid;
  }
}

// ------- Stage 2a: fused IoU-suppression + per-block argmax partials -------
// Each element is owned by exactly one thread per pass, so suppression and
// the post-suppression argmax accumulate locally with no cross-thread hazard.
__global__ __launch_bounds__(256)
void nms_sweep(const float* __restrict__ hdr, const float4* __restrict__ boxes,
               float* __restrict__ live, float* __restrict__ pbest,
               int* __restrict__ pidx) {
  __shared__ float sv[256];
  __shared__ int si[256];
  const int tid = threadIdx.x;
  const int stride = gridDim.x * 256;

  const float sup = hdr[5];             // 1.0 = a box was selected last round
  const float hx1 = hdr[0], hy1 = hdr[1], hx2 = hdr[2], hy2 = hdr[3];
  const float harea = hdr[4];

  float bv = -__builtin_inff();
  int bi = 0x7fffffff;
  for (int i = blockIdx.x * 256 + tid; i < NBOX; i += stride) {  // ascending
    float v = live[i];
    if (sup != 0.0f && v > -1.0e29f) {
      const float4 b = boxes[i];
      const float ix1 = fmaxf(b.x, hx1);
      const float iy1 = fmaxf(b.y, hy1);
      const float ix2 = fminf(b.z, hx2);
      const float iy2 = fminf(b.w, hy2);
      const float inter = fmaxf(ix2 - ix1, 0.0f) * fmaxf(iy2 - iy1, 0.0f);
      const float area = fmaxf(b.z - b.x, 0.0f) * fmaxf(b.w - b.y, 0.0f);
      const float iou = inter / (area + harea - inter + 1e-9f);
      if (iou > 0.5f) {                 // selected box kills itself too
        v = NEG30;
        live[i] = NEG30;
      }
    }
    if (v > bv) { bv = v; bi = i; }     // strict > keeps lowest index
  }
  sv[tid] = bv;
  si[tid] = bi;
  __syncthreads();
  for (int s = 128; s > 0; s >>= 1) {
    if (tid < s) {
      if (sv[tid + s] > sv[tid] ||
          (sv[tid + s] == sv[tid] && si[tid + s] < si[tid])) {
        sv[tid] = sv[tid + s];
        si[tid] = si[tid + s];
      }
    }
    __syncthreads();
  }
  if (tid == 0) { pbest[blockIdx.x] = sv[0]; pidx[blockIdx.x] = si[0]; }
}

// ---------------- Stage 2b: final reduce + emit slot k ---------------------
__global__ __launch_bounds__(512)
void nms_select(const float* __restrict__ pbest, const int* __restrict__ pidx,
                const float4* __restrict__ boxes, const int* __restrict__ cls,
                float* __restrict__ out, float* __restrict__ hdr, int k) {
  __shared__ float sv[512];
  __shared__ int si[512];
  const int tid = threadIdx.x;
  sv[tid] = pbest[tid];
  si[tid] = pidx[tid];
  __syncthreads();
  for (int s = 256; s > 0; s >>= 1) {
    if (tid < s) {
      if (sv[tid + s] > sv[tid] ||
          (sv[tid + s] == sv[tid] && si[tid + s] < si[tid])) {
        sv[tid] = sv[tid + s];
        si[tid] = si[tid + s];
      }
    }
    __syncthreads();
  }
  if (tid == 0) {
    const float bs = sv[0];
    const int bi = si[0];
    const bool ok = bs > NEG30;
    float4 b = make_float4(0.f, 0.f, 0.f, 0.f);
    float cid = -1.0f;
    if (ok) { b = boxes[bi]; cid = (float)cls[bi]; }
    // out: boxes[0:400] | cls[400:500] | scores[500:600] | valid[600:700]
    out[k * 4 + 0] = b.x;
    out[k * 4 + 1] = b.y;
    out[k * 4 + 2] = b.z;
    out[k * 4 + 3] = b.w;
    out[400 + k] = cid;
    out[500 + k] = ok ? bs : 0.0f;
    out[600 + k] = ok ? 1.0f : 0.0f;
    // header for the next fused suppress+argmax pass
    hdr[0] = b.x; hdr[1] = b.y; hdr[2] = b.z; hdr[3] = b.w;
    hdr[4] = fmaxf(b.z - b.x, 0.0f) * fmaxf(b.w - b.y, 0.0f);   // area_i
    hdr[5] = ok ? 1.0f : 0.0f;
  }
}

// ---------------- host-side orchestration ----------------------------------
extern "C" void kernel_launch(void* const* d_in, const int* in_sizes, int n_in,
                              void* d_out, int out_size, void* d_ws, size_t ws_size,
                              hipStream_t stream) {
  (void)in_sizes; (void)n_in; (void)out_size; (void)ws_size;

  const float* in = (const float*)d_in[0];
  const int* sq = (const int*)d_in[1];
  float* out = (float*)d_out;

  // workspace layout (floats): boxes[4N] | live[N] | cls[N] | pbest[512] |
  //                            pidx[512] | hdr[8]   (~12.6 MB total)
  float* ws = (float*)d_ws;
  float4* boxes = (float4*)ws;
  float* live = ws + (size_t)NBOX * 4;
  int* cls = (int*)(ws + (size_t)NBOX * 5);
  float* pbest = ws + (size_t)NBOX * 6;
  int* pidx = (int*)(pbest + 512);
  float* hdr = pbest + 1024;

  decode_kernel<<<CELLS / TILE_CELLS, TILE_CELLS, 0, stream>>>(in, sq, boxes,
                                                               live, cls, hdr);
  // iteration k: suppress with box(k-1) (no-op for k=0), argmax, select k
  for (int k = 0; k < MAX_OUT; ++k) {
    nms_sweep<<<512, 256, 0, stream>>>(hdr, boxes, live, pbest, pidx);
    nms_select<<<1, 512, 0, stream>>>(pbest, pidx, boxes, cls, out, hdr, k);
  }
}